// RelModel_1357209665656
// MI455X (gfx1250) — compile-verified
//
#include <hip/hip_runtime.h>
#include <hip/hip_bf16.h>

// ---------------------------------------------------------------------------
// Problem shapes
// ---------------------------------------------------------------------------
#define N_BOX   512      // N
#define N_CLS   151      // C
#define N_HID   256      // H (K dim of GEMM)
#define N_PAD   152      // padded class stride in LDS (keeps float4 alignment)
#define N_CPAD  160      // padded class count for GEMM tiles (10 * 16)
#define TILES_N 10       // 160/16
#define TILES_M 32       // 512/16

typedef __attribute__((ext_vector_type(2))) float v2f;
typedef __attribute__((ext_vector_type(8))) float v8f;

// ---------------------------------------------------------------------------
// Kernel 0: transpose + zero-pad W_cls [256,151] -> Wt [160][256] (col-major).
// One-time 160KB write into d_ws; makes the GEMM K-loop branch-free and lets
// each lane fetch its B fragment (k, k+1) with a single global_load_b64.
// ---------------------------------------------------------------------------
__global__ __launch_bounds__(256) void relmodel_padw(
    const float* __restrict__ W,   // [256,151]
    float* __restrict__ Wt)        // [160,256]  Wt[c*256+k] = W[k*151+c]
{
    const int idx = blockIdx.x * 256 + threadIdx.x;   // 0 .. 160*256-1
    const int c = idx >> 8;          // / 256
    const int k = idx & 255;         // % 256
    Wt[idx] = (c < N_CLS) ? W[k * N_CLS + c] : 0.0f;
}

// ---------------------------------------------------------------------------
// Kernel 1: obj_dists = ctx_feat @ W_cls + b_cls + prior_dists   (FP32 WMMA)
//
// One wave32 per 16x16 C tile using V_WMMA_F32_16X16X4_F32 (K step = 4).
// A fragment (16x4 f32): lane m=lane%16 holds {A[m][ka], A[m][ka+1]},
// hi = lane/16, ka = k0+2*hi (ISA 32-bit A-matrix layout).
// B fragment (4x16 f32): lane n=lane%16 holds {Wt[n][ka], Wt[n][ka+1]} —
// contiguous thanks to the transposed/padded Wt, so a single b64 load.
// C/D layout: VGPR v -> row = v + 8*hi, col = lane%16.
// Hot loop: 2x global_load_b64 + 1x v_wmma_f32_16x16x4_f32, no branches.
// ---------------------------------------------------------------------------
__global__ __launch_bounds__(128) void relmodel_gemm_wmma(
    const float* __restrict__ A,      // [512,256] ctx_feat
    const float* __restrict__ Wt,     // [160,256] padded/transposed W_cls
    const float* __restrict__ bias,   // [151]     b_cls
    const float* __restrict__ prior,  // [512,151] prior_dists
    float* __restrict__ out)          // [512,151] obj_dists
{
    const int wave = threadIdx.x >> 5;
    const int lane = threadIdx.x & 31;
    const int tile = blockIdx.x * 4 + wave;      // 0..319
    const int tm   = tile / TILES_N;
    const int tn   = tile % TILES_N;
    const int l16  = lane & 15;
    const int hi   = lane >> 4;

    const int row = tm * 16 + l16;               // A row for this lane
    const int col = tn * 16 + l16;               // B/C column for this lane

    const float2* ap = (const float2*)(A  + row * N_HID + 2 * hi); // step 2/iter
    const float2* bp = (const float2*)(Wt + col * N_HID + 2 * hi);

    v8f acc = {};
    #pragma unroll 4
    for (int k4 = 0; k4 < N_HID / 4; ++k4) {     // K step of 4 per WMMA
        const float2 af = ap[k4 * 2];
        const float2 bf = bp[k4 * 2];
        v2f a, b;
        a.x = af.x; a.y = af.y;
        b.x = bf.x; b.y = bf.y;
        // (neg_a, A, neg_b, B, c_mod, C, reuse_a, reuse_b)
        acc = __builtin_amdgcn_wmma_f32_16x16x4_f32(
            false, a, false, b, (short)0, acc, false, false);
    }

    if (col < N_CLS) {
        const float bc = bias[col];
        #pragma unroll
        for (int v = 0; v < 8; ++v) {
            const int r = tm * 16 + v + 8 * hi;
            out[r * N_CLS + col] = acc[v] + bc + prior[r * N_CLS + col];
        }
    }
}

// ---------------------------------------------------------------------------
// Kernel 2: greedy per-class NMS label loop.
//
// Single workgroup, 1024 threads (32 wave32 waves). The whole 512x152 score
// matrix (~304 KB) lives in LDS for the entire 512-iteration greedy loop —
// this fits ONLY because CDNA5 gives 320 KB LDS per WGP. The reference's
// 39.6 MB [N,N,C] overlap tensor is never materialized: each iteration we
// recompute the 512 IoUs of the chosen class lazily (boxes stay L2-resident).
//
// Per iteration:
//   1) vectorized LDS argmax scan (float4 / ds_load_b128, 19 iters/thread)
//   2) wave32 shuffle tree reduction + cross-wave LDS reduction (tie-break
//      on lowest flat index, matching jnp.argmax)
//   3) lazy IoU suppression of the chosen class column
//   4) chosen row := -1.0
// ---------------------------------------------------------------------------
__global__ __launch_bounds__(1024) void relmodel_greedy_nms(
    const float* __restrict__ obj,     // [512,151] obj_dists (from kernel 1)
    const float* __restrict__ boxes,   // [512,151,4] boxes_per_cls
    int* __restrict__ labels)          // [512] int32 labels
{
    extern __shared__ float smem[];
    float* scores = smem;                            // 512*152 floats
    float* rval   = smem + N_BOX * N_PAD;            // 32 floats
    int*   ridx   = (int*)(rval + 32);               // 32 ints
    int*   bcast  = ridx + 32;                       // 1 int

    const int tid = threadIdx.x;
    const int lane = tid & 31;
    const int wid  = tid >> 5;

    // ---- Phase 0: row-wise softmax into LDS; scores[:,0]=0; pad col = -inf
    if (tid < N_BOX) {
        const float* r = obj + tid * N_CLS;
        float m = r[0];
        for (int c = 1; c < N_CLS; ++c) m = fmaxf(m, r[c]);
        float s = 0.0f;
        for (int c = 0; c < N_CLS; ++c) s += __expf(r[c] - m);
        const float inv = 1.0f / s;
        float* srow = scores + tid * N_PAD;
        srow[0] = 0.0f;
        for (int c = 1; c < N_CLS; ++c) srow[c] = __expf(r[c] - m) * inv;
        srow[N_CLS] = -1e30f;            // padding column never wins argmax
        labels[tid] = 0;                 // jnp.zeros default
    }
    __syncthreads();

    const float4* bp = (const float4*)boxes;         // [512*151] float4 boxes

    for (int it = 0; it < N_BOX; ++it) {
        // ---- 1) strided float4 scan of LDS scores (first-index tie keep)
        float bv = -3.4e38f;
        int   bi = 0x7fffffff;
        const float4* s4 = (const float4*)scores;
        #pragma unroll 4
        for (int j = tid; j < (N_BOX * N_PAD) / 4; j += 1024) {
            const float4 q = s4[j];
            const int base = j * 4;
            if (q.x > bv) { bv = q.x; bi = base;     }
            if (q.y > bv) { bv = q.y; bi = base + 1; }
            if (q.z > bv) { bv = q.z; bi = base + 2; }
            if (q.w > bv) { bv = q.w; bi = base + 3; }
        }
        // ---- 2) wave32 shuffle reduction (tie-break: lowest flat index)
        #pragma unroll
        for (int off = 16; off > 0; off >>= 1) {
            const float ov = __shfl_down(bv, off, 32);
            const int   oi = __shfl_down(bi, off, 32);
            if (ov > bv || (ov == bv && oi < bi)) { bv = ov; bi = oi; }
        }
        if (lane == 0) { rval[wid] = bv; ridx[wid] = bi; }
        __syncthreads();
        if (tid < 32) {
            bv = rval[tid]; bi = ridx[tid];
            #pragma unroll
            for (int off = 16; off > 0; off >>= 1) {
                const float ov = __shfl_down(bv, off, 32);
                const int   oi = __shfl_down(bi, off, 32);
                if (ov > bv || (ov == bv && oi < bi)) { bv = ov; bi = oi; }
            }
            if (tid == 0) {
                bcast[0] = bi;
                labels[bi / N_PAD] = bi % N_PAD;     // pad col ordering == ref
            }
        }
        __syncthreads();

        const int best = bcast[0];
        const int brow = best / N_PAD;
        const int bcls = best - brow * N_PAD;

        // ---- 3) lazy IoU suppression of the chosen class column
        if (tid < N_BOX) {
            const float4 ca = bp[brow * N_CLS + bcls];   // chosen box (L2 hit)
            const float4 cb = bp[tid  * N_CLS + bcls];   // this row's box
            const float x1 = fmaxf(ca.x, cb.x);
            const float y1 = fmaxf(ca.y, cb.y);
            const float x2 = fminf(ca.z, cb.z);
            const float y2 = fminf(ca.w, cb.w);
            const float inter = fmaxf(x2 - x1, 0.0f) * fmaxf(y2 - y1, 0.0f);
            const float areaA = (ca.z - ca.x) * (ca.w - ca.y);
            const float areaB = (cb.z - cb.x) * (cb.w - cb.y);
            const float uni   = areaA + areaB - inter;
            const float iou   = inter / fmaxf(uni, 1e-8f);
            if (iou >= 0.3f) scores[tid * N_PAD + bcls] = 0.0f;
        }
        __syncthreads();
        // ---- 4) chosen row suppressed (wins over the 0.0 write above)
        if (tid < N_CLS) scores[brow * N_PAD + tid] = -1.0f;
        __syncthreads();
    }
}

// ---------------------------------------------------------------------------
// Harness entry point
// ---------------------------------------------------------------------------
extern "C" void kernel_launch(void* const* d_in, const int* in_sizes, int n_in,
                              void* d_out, int out_size, void* d_ws, size_t ws_size,
                              hipStream_t stream) {
    const float* ctx   = (const float*)d_in[0];   // [512,256]
    const float* prior = (const float*)d_in[1];   // [512,151]
    const float* boxes = (const float*)d_in[2];   // [512,151,4]
    const float* W     = (const float*)d_in[3];   // [256,151]
    const float* bcls  = (const float*)d_in[4];   // [151]

    float* obj = (float*)d_out;                          // [512,151]
    int*   lab = (int*)d_out + N_BOX * N_CLS;            // [512] int32
    float* Wt  = (float*)d_ws;                           // [160,256] scratch

    // 0) transpose + zero-pad W (160*256 elements, 160 blocks x 256 threads)
    relmodel_padw<<<(N_CPAD * N_HID) / 256, 256, 0, stream>>>(W, Wt);

    // 1) 320 tiles, 4 waves (one tile each) per 128-thread block
    relmodel_gemm_wmma<<<(TILES_M * TILES_N) / 4, 128, 0, stream>>>(
        ctx, Wt, bcls, prior, obj);

    // 2) single workgroup owning the entire score matrix in 320KB CDNA5 LDS
    const size_t shmem = (size_t)(N_BOX * N_PAD) * sizeof(float)  // 311296 B
                       + 32 * sizeof(float) + 32 * sizeof(int) + 16;
    relmodel_greedy_nms<<<1, 1024, shmem, stream>>>(obj, boxes, lab);
}